// GTE_Attention_65687229825299
// MI455X (gfx1250) — compile-verified
//
#include <hip/hip_runtime.h>
#include <hip/hip_bf16.h>
#include <math.h>

#define N_NODES 50000
#define N_EDGES 500000
#define IN_DIM  128
#define HEADS   8
#define HDIM    16
#define HD      (HEADS*HDIM)      // 128
#define CLAMP_V 5.0f

typedef float v2f __attribute__((ext_vector_type(2)));
typedef float v8f __attribute__((ext_vector_type(8)));

// ---- order-preserving float<->uint key for atomicMax-based segment_max ----
__device__ __forceinline__ unsigned encKey(float f) {
    unsigned u = __float_as_uint(f);
    return (u & 0x80000000u) ? ~u : (u | 0x80000000u);
}
__device__ __forceinline__ float decKey(unsigned k) {
    unsigned u = (k & 0x80000000u) ? (k & 0x7FFFFFFFu) : ~k;
    return __uint_as_float(u);
}

// ============================ P0: init ============================
__global__ __launch_bounds__(256) void k_init(float* __restrict__ wV,
                                              float* __restrict__ denom,
                                              unsigned* __restrict__ smax) {
    int i = blockIdx.x * 256 + threadIdx.x;
    if (i < N_NODES * HD) wV[i] = 0.0f;
    if (i < N_NODES * HEADS) {
        denom[i] = 0.0f;
        smax[i]  = encKey(-3.0e38f);
    }
}

// ============================ P1: node QKV GEMM ============================
// 16x16 output tile per wave via V_WMMA_F32_16X16X4_F32 (K=4, 32 steps).
// A frag (16x4 f32): lanes 0-15 hold K={kb,kb+1}, lanes 16-31 hold K={kb+2,kb+3}, M=lane%16.
// B frag mirrors with N=lane%16. C: VGPR r -> row r (lanes<16) / row 8+r (lanes>=16), col=lane%16.
__global__ __launch_bounds__(256) void k_qkv(const float* __restrict__ x,
                                             const float* __restrict__ QW,
                                             const float* __restrict__ Qb,
                                             const float* __restrict__ KW,
                                             const float* __restrict__ VW,
                                             float* __restrict__ Qh,
                                             float* __restrict__ Kh,
                                             float* __restrict__ Vh) {
    const int wave   = blockIdx.x * 8 + (threadIdx.x >> 5);   // 75000 waves exactly
    const int rowT   = wave / 24;
    const int ct     = wave % 24;
    const int lane   = threadIdx.x & 31;
    const int half   = lane >> 4;
    const int l      = lane & 15;

    const float* W;  const float* bias = nullptr;  float* out;  int c0;
    if (ct < 8)       { W = QW; bias = Qb; out = Qh; c0 = ct * 16; }
    else if (ct < 16) { W = KW;            out = Kh; c0 = (ct - 8) * 16; }
    else              { W = VW;            out = Vh; c0 = (ct - 16) * 16; }

    const int row0 = rowT * 16;
    const float bv = bias ? bias[c0 + l] : 0.0f;
    v8f acc = {bv, bv, bv, bv, bv, bv, bv, bv};

    const float* arow = x + (row0 + l) * IN_DIM + 2 * half;   // K base per half-wave
    const float* bcol = W + (2 * half) * HD + c0 + l;         // ldb = 128

    __builtin_prefetch(arow);
    #pragma unroll 4
    for (int kk = 0; kk < 32; ++kk) {
        v2f a = *(const v2f*)(arow + kk * 4);
        v2f b;
        b.x = bcol[kk * 4 * HD];
        b.y = bcol[kk * 4 * HD + HD];
        acc = __builtin_amdgcn_wmma_f32_16x16x4_f32(false, a, false, b,
                                                    (short)0, acc, false, false);
    }

    float* o = out + (row0 + 8 * half) * HD + c0 + l;
    #pragma unroll
    for (int r = 0; r < 8; ++r) o[r * HD] = acc[r];
}

// ============================ P2: edge GEMM + score ============================
// One wave per (16-edge tile, head). Two accumulators: E_w (cols h*32..+16) and
// E_b (cols h*32+16..+32) of Ep = edge_attr @ EW + Eb.  Fused epilogue computes
// score = K[src]*Q[dst]*E_w + E_b, writes wE, row-reduces to s, atomicMax smax.
__global__ __launch_bounds__(256) void k_edge(const float* __restrict__ edge_attr,
                                              const int* __restrict__ eidx,   // [2][E]
                                              const float* __restrict__ EW,   // 128x256
                                              const float* __restrict__ Eb,   // 256
                                              const float* __restrict__ Qh,
                                              const float* __restrict__ Kh,
                                              float* __restrict__ wE,
                                              float* __restrict__ s_ws,
                                              unsigned* __restrict__ smax) {
    const int wave = blockIdx.x * 8 + (threadIdx.x >> 5);    // 250000 waves exactly
    const int tile = wave >> 3;
    const int h    = wave & 7;
    const int lane = threadIdx.x & 31;
    const int half = lane >> 4;
    const int l    = lane & 15;

    const int row0 = tile * 16;
    const int c0w  = h * 32;
    const int c0b  = h * 32 + 16;

    const float bw = Eb[c0w + l];
    const float bb = Eb[c0b + l];
    v8f accw = {bw, bw, bw, bw, bw, bw, bw, bw};
    v8f accb = {bb, bb, bb, bb, bb, bb, bb, bb};

    const float* arow = edge_attr + (row0 + l) * IN_DIM + 2 * half;
    const float* bw0  = EW + (2 * half) * 256 + c0w + l;     // ldb = 256
    const float* bb0  = EW + (2 * half) * 256 + c0b + l;

    __builtin_prefetch(arow);
    #pragma unroll 4
    for (int kk = 0; kk < 32; ++kk) {
        v2f a = *(const v2f*)(arow + kk * 4);
        v2f b1, b2;
        b1.x = bw0[kk * 4 * 256];
        b1.y = bw0[kk * 4 * 256 + 256];
        b2.x = bb0[kk * 4 * 256];
        b2.y = bb0[kk * 4 * 256 + 256];
        accw = __builtin_amdgcn_wmma_f32_16x16x4_f32(false, a, false, b1,
                                                     (short)0, accw, false, false);
        accb = __builtin_amdgcn_wmma_f32_16x16x4_f32(false, a, false, b2,
                                                     (short)0, accb, false, false);
    }

    const int* src = eidx;
    const int* dst = eidx + N_EDGES;

    #pragma unroll
    for (int r = 0; r < 8; ++r) {
        const int e   = row0 + 8 * half + r;      // C-layout row for this half-wave
        const int sE  = src[e];
        const int dE  = dst[e];
        const float kv = Kh[sE * HD + h * HDIM + l];
        const float qv = Qh[dE * HD + h * HDIM + l];
        float sc = kv * qv * accw[r] + accb[r];
        wE[e * HD + h * HDIM + l] = sc;           // wE == score (e_t)

        // reduce over the 16 feature lanes of this half-wave
        float red = sc;
        red += __shfl_xor(red, 1, 16);
        red += __shfl_xor(red, 2, 16);
        red += __shfl_xor(red, 4, 16);
        red += __shfl_xor(red, 8, 16);
        if (l == 0) {
            float s = red * 0.25f;                // 1/sqrt(16)
            s = fminf(CLAMP_V, fmaxf(-CLAMP_V, s));
            s_ws[e * HEADS + h] = s;
            atomicMax(&smax[dE * HEADS + h], encKey(s));
        }
    }
}

// ============================ P3: exp + denom ============================
__global__ __launch_bounds__(256) void k_expden(const int* __restrict__ eidx,
                                                const float* __restrict__ s_ws,
                                                const unsigned* __restrict__ smax,
                                                float* __restrict__ ex_ws,
                                                float* __restrict__ denom) {
    int i = blockIdx.x * 256 + threadIdx.x;
    if (i >= N_EDGES * HEADS) return;
    const int e = i >> 3;
    const int h = i & 7;
    const int dE = eidx[N_EDGES + e];
    const float m  = decKey(smax[dE * HEADS + h]);
    const float ex = expf(s_ws[i] - m);
    ex_ws[i] = ex;
    atomicAdd(&denom[dE * HEADS + h], ex);
}

// ============================ P4: normalize + scatter wV ============================
// wV[dst] += (V[src] + e_t) * attn, float4 per thread, f32 atomics.
__global__ __launch_bounds__(256) void k_scatter(const int* __restrict__ eidx,
                                                 const float* __restrict__ ex_ws,
                                                 const float* __restrict__ denom,
                                                 const float* __restrict__ Vh,
                                                 const float* __restrict__ wE,
                                                 float* __restrict__ wV) {
    int i = blockIdx.x * 256 + threadIdx.x;    // E * 32 threads
    if (i >= N_EDGES * 32) return;
    const int e  = i >> 5;
    const int rm = i & 31;
    const int h  = rm >> 2;
    const int d0 = (rm & 3) * 4;

    const int sE = eidx[e];
    const int dE = eidx[N_EDGES + e];
    const float attn = ex_ws[e * HEADS + h] / (denom[dE * HEADS + h] + 1e-16f);

    const float4 v = *(const float4*)(Vh + sE * HD + h * HDIM + d0);
    const float4 t = *(const float4*)(wE + e * HD + h * HDIM + d0);
    float* o = wV + dE * HD + h * HDIM + d0;
    atomicAdd(o + 0, (v.x + t.x) * attn);
    atomicAdd(o + 1, (v.y + t.y) * attn);
    atomicAdd(o + 2, (v.z + t.z) * attn);
    atomicAdd(o + 3, (v.w + t.w) * attn);
}

// ============================ launcher ============================
extern "C" void kernel_launch(void* const* d_in, const int* in_sizes, int n_in,
                              void* d_out, int out_size, void* d_ws, size_t ws_size,
                              hipStream_t stream) {
    (void)in_sizes; (void)n_in; (void)out_size; (void)ws_size;

    const float* x         = (const float*)d_in[0];
    const float* edge_attr = (const float*)d_in[1];
    const int*   eidx      = (const int*)d_in[2];
    const float* QW        = (const float*)d_in[3];
    const float* Qb        = (const float*)d_in[4];
    const float* KW        = (const float*)d_in[5];
    const float* EW        = (const float*)d_in[6];
    const float* Eb        = (const float*)d_in[7];
    const float* VW        = (const float*)d_in[8];

    float* out  = (float*)d_out;
    float* wV   = out;                       // N*128
    float* wE   = out + N_NODES * HD;        // E*128

    // workspace layout (floats)
    float* w      = (float*)d_ws;
    float* Qh     = w;                                   // 6.4M
    float* Kh     = Qh + N_NODES * HD;                   // 6.4M
    float* Vh     = Kh + N_NODES * HD;                   // 6.4M
    float* s_ws   = Vh + N_NODES * HD;                   // 4.0M
    float* ex_ws  = s_ws + N_EDGES * HEADS;              // 4.0M
    float* denom  = ex_ws + N_EDGES * HEADS;             // 0.4M
    unsigned* smx = (unsigned*)(denom + N_NODES * HEADS);// 0.4M

    // P0: init accumulators
    k_init<<<(N_NODES * HD + 255) / 256, 256, 0, stream>>>(wV, denom, smx);

    // P1: Q/K/V node GEMM  (3125 row tiles * 24 col tiles = 75000 waves, 8 waves/block)
    k_qkv<<<75000 / 8, 256, 0, stream>>>(x, QW, Qb, KW, VW, Qh, Kh, Vh);

    // P2: edge GEMM + score + wE + segment max  (31250 tiles * 8 heads = 250000 waves)
    k_edge<<<250000 / 8, 256, 0, stream>>>(edge_attr, eidx, EW, Eb, Qh, Kh,
                                           wE, s_ws, smx);

    // P3: softmax exp + denominator
    k_expden<<<(N_EDGES * HEADS + 255) / 256, 256, 0, stream>>>(eidx, s_ws, smx,
                                                                ex_ws, denom);

    // P4: normalize + scatter-add into wV
    k_scatter<<<(N_EDGES * 32 + 255) / 256, 256, 0, stream>>>(eidx, ex_ws, denom,
                                                              Vh, wE, wV);
}